// CAE_21242908246023
// MI455X (gfx1250) — compile-verified
//
#include <hip/hip_runtime.h>
#include <hip/hip_bf16.h>

typedef __attribute__((ext_vector_type(2))) float v2f;
typedef __attribute__((ext_vector_type(8))) float v8f;

#define NB 16384
#define ND 5000
#define NL 64
#define NH 192          // 3*L concatenated hidden
#define DEC_SCALE 0.0159f

// ---------------------------------------------------------------------------
// Prep: build concatenated encoder weight We[k][n] (k=0..D-1, n=0..191)
//   n<64   -> W_base[n, k]      (W_base is [L,D] row-major)
//   n<128  -> W_enc_batch[n-64, k]
//   else   -> W_enc_cell[n-128, k]
// ---------------------------------------------------------------------------
__global__ __launch_bounds__(256) void build_We_kernel(
    const float* __restrict__ Wb, const float* __restrict__ Web,
    const float* __restrict__ Wec, float* __restrict__ We) {
  int e = blockIdx.x * 256 + threadIdx.x;
  if (e >= ND * NH) return;
  int n = e % NH;
  int k = e / NH;
  float v;
  if (n < 64)       v = Wb [(size_t)n         * ND + k];
  else if (n < 128) v = Web[(size_t)(n - 64)  * ND + k];
  else              v = Wec[(size_t)(n - 128) * ND + k];
  We[e] = v;
}

// ---------------------------------------------------------------------------
// Prep: build concatenated decoder weight Wc[l][d] (l=0..191, d=0..D-1)
//   l<64   -> W_base[l, d]              ([L,D] row-major, direct copy)
//   l<128  -> W_dec_batch[d, l-64]      ([D,L] row-major, transpose)
//   else   -> W_dec_cell[d, l-128]
// ---------------------------------------------------------------------------
__global__ __launch_bounds__(256) void build_Wc_kernel(
    const float* __restrict__ Wb, const float* __restrict__ Wdb,
    const float* __restrict__ Wdc, float* __restrict__ Wc) {
  int e = blockIdx.x * 256 + threadIdx.x;
  if (e >= NH * ND) return;
  int d = e % ND;
  int l = e / ND;
  float v;
  if (l < 64)       v = Wb [(size_t)l * ND + d];
  else if (l < 128) v = Wdb[(size_t)d * 64 + (l - 64)];
  else              v = Wdc[(size_t)d * 64 + (l - 128)];
  Wc[e] = v;
}

// ---------------------------------------------------------------------------
// f32 WMMA GEMM: C[M,N] = A[M,K] @ B[K,N], all row-major, ldA=K, ldB=N, ldC=N
// Workgroup: 256 threads = 8 wave32; 64 output rows per WG.
// Wave w -> row tile (w>>1)*16, col group (w&1) covering (BLK_N/2) columns.
// LDS layouts chosen so each WMMA fragment is one aligned ds_load_b64:
//   LA row-major, stride 36 dwords (16B-aligned rows, conflict-free frags)
//   LB n-major,   stride 34 dwords ((k,k+1) contiguous, 8B aligned)
// ---------------------------------------------------------------------------
template <int BLK_N>
__global__ __launch_bounds__(256) void gemm_f32_wmma(
    const float* __restrict__ A, const float* __restrict__ Bm,
    float* __restrict__ C, int M, int N, int K) {
  constexpr int KC  = 32;
  constexpr int LAS = KC + 4;             // LA stride (dwords), mult of 4
  constexpr int LBS = KC + 2;             // LB stride (dwords), even
  constexpr int CPW = (BLK_N / 16) / 2;   // col tiles per wave

  __shared__ __align__(16) float LA[64 * LAS];     // [row][k]
  __shared__ __align__(16) float LB[BLK_N * LBS];  // [n][k]

  const int tid   = threadIdx.x;
  const int lane  = tid & 31;
  const int wave  = tid >> 5;
  const int rt    = wave >> 1;            // row tile 0..3
  const int cg    = wave & 1;             // col group 0..1
  const int r0    = blockIdx.x * 64;
  const int n0    = blockIdx.y * BLK_N;
  const int mrow  = lane & 15;
  const int khalf = lane >> 4;            // 0 for lanes 0-15, 1 for 16-31

  v8f acc[CPW];
#pragma unroll
  for (int c = 0; c < CPW; ++c)
    acc[c] = (v8f){0.f, 0.f, 0.f, 0.f, 0.f, 0.f, 0.f, 0.f};

  const float* pa = &LA[(rt * 16 + mrow) * LAS + 2 * khalf];
  const int    nb = cg * (CPW * 16) + mrow;   // wave's first B column (of BLK_N)

  for (int kc = 0; kc < K; kc += KC) {
    __syncthreads();
    // ---- Stage A chunk: 64 x 32, float4 global loads, b128 LDS stores ----
#pragma unroll
    for (int i = 0; i < (64 * KC) / (4 * 256); ++i) {
      int e4   = tid + i * 256;           // 0..511
      int row  = e4 >> 3;                 // KC/4 = 8 float4 per row
      int col4 = (e4 & 7) * 4;
      int gk   = kc + col4;
      const float* src = A + (size_t)(r0 + row) * K + gk;
      float4 v;
      if (gk + 3 < K) {
        v = *(const float4*)src;
      } else {
        v.x = (gk     < K) ? src[0] : 0.f;
        v.y = (gk + 1 < K) ? src[1] : 0.f;
        v.z = (gk + 2 < K) ? src[2] : 0.f;
        v.w = (gk + 3 < K) ? src[3] : 0.f;
      }
      *(float4*)&LA[row * LAS + col4] = v;
    }
    // ---- Stage B chunk: 32 x BLK_N, float4 global loads, transposed to LDS ----
#pragma unroll
    for (int i = 0; i < (KC * BLK_N) / (4 * 256); ++i) {
      int e4   = tid + i * 256;
      int col4 = (e4 % (BLK_N / 4)) * 4;
      int row  = e4 / (BLK_N / 4);        // k within chunk
      int gk   = kc + row;
      int gn   = n0 + col4;
      const float* src = Bm + (size_t)gk * N + gn;
      float4 v;
      if (gk < K && gn + 3 < N) {
        v = *(const float4*)src;
      } else {
        bool kok = gk < K;
        v.x = (kok && gn     < N) ? src[0] : 0.f;
        v.y = (kok && gn + 1 < N) ? src[1] : 0.f;
        v.z = (kok && gn + 2 < N) ? src[2] : 0.f;
        v.w = (kok && gn + 3 < N) ? src[3] : 0.f;
      }
      LB[(col4 + 0) * LBS + row] = v.x;
      LB[(col4 + 1) * LBS + row] = v.y;
      LB[(col4 + 2) * LBS + row] = v.z;
      LB[(col4 + 3) * LBS + row] = v.w;
    }
    __syncthreads();

#pragma unroll
    for (int kk = 0; kk < KC; kk += 4) {
      // A fragment 16x4 f32: VGPR0 = K0(lanes0-15)/K2(lanes16-31), VGPR1 = K1/K3
      v2f a = *(const v2f*)(pa + kk);     // 8B aligned: LAS mult of 4, kk+2h even
#pragma unroll
      for (int c = 0; c < CPW; ++c) {
        // B fragment 4x16 f32: VGPR0 = rows K0/K2 across lane halves, VGPR1 = K1/K3
        v2f b = *(const v2f*)(&LB[(nb + c * 16) * LBS + kk + 2 * khalf]);
        acc[c] = __builtin_amdgcn_wmma_f32_16x16x4_f32(
            false, a, false, b, (short)0, acc[c], false, false);
      }
    }
  }

  // Store C: VGPR r holds rows (rbase+r) lanes 0-15 and (rbase+8+r) lanes 16-31
#pragma unroll
  for (int c = 0; c < CPW; ++c) {
    int ncol = n0 + cg * (CPW * 16) + c * 16 + mrow;
    if (ncol < N) {
      int rbase = r0 + rt * 16 + (khalf ? 8 : 0);
#pragma unroll
      for (int r = 0; r < 8; ++r)
        C[(size_t)(rbase + r) * N + ncol] = acc[c][r];
    }
  }
}

// ---------------------------------------------------------------------------
// Heads: per row b, for each modality (offset 64: batch, offset 128: cell):
//   x    = H[b, off:off+64]
//   h_f  = W_heads[src[b]] @ x          (y_m = sum_l W[ctx,m,l] * x_l)
//   dec  = W_heads[tgt[b]] @ h_f
//   H[b, off:off+64] = DEC_SCALE * dec   (in place; cols 0..63 = h_base kept)
// One wave32 per row; lane owns outputs m=lane and m=lane+32; weight rows are
// 256B-aligned -> float4 loads; x broadcast via wave shuffles.
// ---------------------------------------------------------------------------
__device__ __forceinline__ void head_matvec(const float* __restrict__ Wr,
                                            int lane, float x0, float x1,
                                            float& y0, float& y1) {
  const float4* w0 = (const float4*)(Wr + (size_t)lane * 64);
  const float4* w1 = (const float4*)(Wr + (size_t)(lane + 32) * 64);
  float a0 = 0.f, a1 = 0.f;
#pragma unroll
  for (int q = 0; q < 8; ++q) {           // l = 0..31 broadcast from x0
    float4 u0 = w0[q], u1 = w1[q];
    int l = q * 4;
    float xa = __shfl(x0, l,     32);
    float xb = __shfl(x0, l + 1, 32);
    float xc = __shfl(x0, l + 2, 32);
    float xd = __shfl(x0, l + 3, 32);
    a0 += u0.x * xa + u0.y * xb + u0.z * xc + u0.w * xd;
    a1 += u1.x * xa + u1.y * xb + u1.z * xc + u1.w * xd;
  }
#pragma unroll
  for (int q = 8; q < 16; ++q) {          // l = 32..63 broadcast from x1
    float4 u0 = w0[q], u1 = w1[q];
    int l = (q - 8) * 4;
    float xa = __shfl(x1, l,     32);
    float xb = __shfl(x1, l + 1, 32);
    float xc = __shfl(x1, l + 2, 32);
    float xd = __shfl(x1, l + 3, 32);
    a0 += u0.x * xa + u0.y * xb + u0.z * xc + u0.w * xd;
    a1 += u1.x * xa + u1.y * xb + u1.z * xc + u1.w * xd;
  }
  y0 = a0;
  y1 = a1;
}

__global__ __launch_bounds__(256) void heads_kernel(
    float* __restrict__ H,
    const int* __restrict__ src_b, const int* __restrict__ tgt_b,
    const int* __restrict__ src_c, const int* __restrict__ tgt_c,
    const float* __restrict__ Whb, const float* __restrict__ Whc) {
  const int lane = threadIdx.x & 31;
  const int wave = threadIdx.x >> 5;
  const int b = blockIdx.x * 8 + wave;
  if (b >= NB) return;
  float* row = H + (size_t)b * NH;

  // --- batch modality ---
  {
    float x0 = row[64 + lane];
    float x1 = row[64 + 32 + lane];
    const float* Ws = Whb + (size_t)src_b[b] * 4096;
    const float* Wt = Whb + (size_t)tgt_b[b] * 4096;
    float h0, h1, d0, d1;
    head_matvec(Ws, lane, x0, x1, h0, h1);
    head_matvec(Wt, lane, h0, h1, d0, d1);
    row[64 + lane]      = DEC_SCALE * d0;
    row[64 + 32 + lane] = DEC_SCALE * d1;
  }
  // --- cell modality ---
  {
    float x0 = row[128 + lane];
    float x1 = row[128 + 32 + lane];
    const float* Ws = Whc + (size_t)src_c[b] * 4096;
    const float* Wt = Whc + (size_t)tgt_c[b] * 4096;
    float h0, h1, d0, d1;
    head_matvec(Ws, lane, x0, x1, h0, h1);
    head_matvec(Wt, lane, h0, h1, d0, d1);
    row[128 + lane]      = DEC_SCALE * d0;
    row[128 + 32 + lane] = DEC_SCALE * d1;
  }
}

// ---------------------------------------------------------------------------
extern "C" void kernel_launch(void* const* d_in, const int* in_sizes, int n_in,
                              void* d_out, int out_size, void* d_ws, size_t ws_size,
                              hipStream_t stream) {
  const float* expr     = (const float*)d_in[0];
  const int*   src_b    = (const int*)d_in[1];
  const int*   tgt_b    = (const int*)d_in[2];
  const int*   src_c    = (const int*)d_in[3];
  const int*   tgt_c    = (const int*)d_in[4];
  const float* W_base   = (const float*)d_in[5];
  const float* W_enc_b  = (const float*)d_in[6];
  const float* W_dec_b  = (const float*)d_in[7];
  const float* W_hd_b   = (const float*)d_in[8];
  const float* W_enc_c  = (const float*)d_in[9];
  const float* W_dec_c  = (const float*)d_in[10];
  const float* W_hd_c   = (const float*)d_in[11];
  float* logits = (float*)d_out;

  // workspace layout (all 16B aligned): We | Wc | H
  const size_t we_bytes = (size_t)ND * NH * sizeof(float);  // 3.84 MB
  const size_t wc_bytes = (size_t)NH * ND * sizeof(float);  // 3.84 MB
  char* ws = (char*)d_ws;
  float* We = (float*)ws;
  float* Wc = (float*)(ws + we_bytes);
  float* H  = (float*)(ws + we_bytes + wc_bytes);           // 16384 x 192

  const int prep_blocks = (ND * NH + 255) / 256;
  build_We_kernel<<<prep_blocks, 256, 0, stream>>>(W_base, W_enc_b, W_enc_c, We);
  build_Wc_kernel<<<prep_blocks, 256, 0, stream>>>(W_base, W_dec_b, W_dec_c, Wc);

  // H[B,192] = expr[B,5000] @ We[5000,192]   (expr read exactly once: grid.y==1)
  gemm_f32_wmma<NH><<<dim3(NB / 64, 1), 256, 0, stream>>>(expr, We, H, NB, NH, ND);

  // per-row head matvecs, scaled, in place
  heads_kernel<<<NB / 8, 256, 0, stream>>>(H, src_b, tgt_b, src_c, tgt_c, W_hd_b, W_hd_c);

  // logits[B,5000] = H[B,192] @ Wc[192,5000]
  gemm_f32_wmma<128><<<dim3(NB / 64, (ND + 127) / 128), 256, 0, stream>>>(
      H, Wc, logits, NB, ND, NH);
}